// ConstLoss_22746146800082
// MI455X (gfx1250) — compile-verified
//
#include <hip/hip_runtime.h>
#include <hip/hip_bf16.h>

// CDNA5 / gfx1250: wave32, WMMA (not MFMA), async global->LDS (ASYNCcnt).
typedef __attribute__((ext_vector_type(2))) float v2f;   // 16x4 f32 A/B fragment: 2 VGPRs/lane
typedef __attribute__((ext_vector_type(8))) float v8f;   // 16x16 f32 C/D fragment: 8 VGPRs/lane

#define BLK 128                 // 128x128 output block per workgroup
#define KB 16                   // K-chunk staged in LDS
#define LDS_STRIDE (KB + 4)     // 20 floats: conflict-free across 16 rows, 80B rows (16B aligned)
#define EPS_NORM 1e-7f
#define EPS_COS 1e-8f

// ---------------------------------------------------------------------------
// Kernel 1: per-row norms. a[i] = 1/(||x_i||+1e-7), s[i] = ||x_i||/(||x_i||+1e-7)
// ---------------------------------------------------------------------------
__global__ __launch_bounds__(256) void row_norms_kernel(const float* __restrict__ x,
                                                        float* __restrict__ a,
                                                        float* __restrict__ s,
                                                        int D) {
    int row  = blockIdx.x * 8 + (threadIdx.x >> 5);
    int lane = threadIdx.x & 31;
    const float* xr = x + (size_t)row * D;
    float acc = 0.0f;
    for (int c = lane * 4; c < D; c += 32 * 4) {
        float4 v = *(const float4*)(xr + c);
        acc += v.x * v.x + v.y * v.y + v.z * v.z + v.w * v.w;
    }
#pragma unroll
    for (int off = 16; off > 0; off >>= 1) acc += __shfl_xor(acc, off, 32);
    if (lane == 0) {
        float n   = sqrtf(acc);
        float inv = 1.0f / (n + EPS_NORM);
        a[row] = inv;
        s[row] = n * inv;
    }
}

// LDS byte offset of a generic pointer to __shared__ memory: per the flat
// aperture layout, the low 32 bits of a generic LDS address are the LDS offset.
__device__ __forceinline__ uint32_t lds_u32(const void* p) {
    return (uint32_t)(uintptr_t)p;
}

// Two GLOBAL_LOAD_ASYNC_TO_LDS_B128 issues: this thread's 8-float slice.
// GV mode (saddr = off): per-lane 64-bit global addr, per-lane LDS dest VGPR.
__device__ __forceinline__ void issue_async_b128x2(const float* gp, uint32_t lo) {
    asm volatile(
        "global_load_async_to_lds_b128 %0, %2, off\n\t"
        "global_load_async_to_lds_b128 %1, %3, off"
        :
        : "v"(lo), "v"(lo + 16u), "v"(gp), "v"(gp + 4)
        : "memory");
}

// ---------------------------------------------------------------------------
// Kernel 2: fused Gram-difference over a triangular grid of 128x128 blocks.
// Async double-buffered LDS staging; per wave 8 WMMA tiles x 2 matrices.
// ---------------------------------------------------------------------------
__global__ __launch_bounds__(256) void gram_diff_kernel(
    const float* __restrict__ Xq, const float* __restrict__ Xk,
    const float* __restrict__ aq, const float* __restrict__ sq,
    const float* __restrict__ ak, const float* __restrict__ sk,
    float* __restrict__ partial, int N, int D, int nblk)
{
    // Decode triangular block index t -> (bi, bj), bi <= bj.
    int t = blockIdx.x;
    int bi = 0, rem = t, rowlen = nblk;
    while (rem >= rowlen) { rem -= rowlen; rowlen--; bi++; }
    int bj = bi + rem;

    // [buffer][operand][row][col]; operands: 0=qI 1=qJ 2=kI 3=kJ.  ~80 KB.
    __shared__ __align__(16) float ldsbuf[2][4][BLK][LDS_STRIDE];
    __shared__ float wave_sums[8];

    const int tid  = threadIdx.x;
    const int lane = tid & 31;
    const int wave = tid >> 5;
    const int hf   = lane >> 4;      // half-wave select
    const int ln   = lane & 15;

    const int Ibase = bi * BLK;
    const int Jbase = bj * BLK;

    v8f accq[8], acck[8];
    const v8f vzero = {0.f, 0.f, 0.f, 0.f, 0.f, 0.f, 0.f, 0.f};
#pragma unroll
    for (int i = 0; i < 8; ++i) { accq[i] = vzero; acck[i] = vzero; }

    // Cooperative staging: each thread owns (row ldr, cols ldc..ldc+7) of every
    // operand tile -> 2 async b128 per operand per chunk, 8 per chunk total.
    const int ldr = tid >> 1;            // 0..127
    const int ldc = (tid & 1) * 8;       // 0 or 8

    const float* gsrc[4];
    gsrc[0] = Xq + (size_t)(Ibase + ldr) * D + ldc;
    gsrc[1] = Xq + (size_t)(Jbase + ldr) * D + ldc;
    gsrc[2] = Xk + (size_t)(Ibase + ldr) * D + ldc;
    gsrc[3] = Xk + (size_t)(Jbase + ldr) * D + ldc;

    uint32_t loff[2][4];
#pragma unroll
    for (int b = 0; b < 2; ++b)
#pragma unroll
        for (int op = 0; op < 4; ++op)
            loff[b][op] = lds_u32(&ldsbuf[b][op][ldr][ldc]);

    const int nchunks = D / KB;

    // Prologue: chunk 0 -> buffer 0.
#pragma unroll
    for (int op = 0; op < 4; ++op) issue_async_b128x2(gsrc[op], loff[0][op]);

    for (int c = 0; c < nchunks; ++c) {
        const int cur = c & 1;
        if (c + 1 < nchunks) {
            const int koff = (c + 1) * KB;
#pragma unroll
            for (int op = 0; op < 4; ++op)
                issue_async_b128x2(gsrc[op] + koff, loff[cur ^ 1][op]);
            // 8 newly issued outstanding; <=8 means the current buffer's 8 are done.
            asm volatile("s_wait_asynccnt 0x8" ::: "memory");
        } else {
            asm volatile("s_wait_asynccnt 0x0" ::: "memory");
        }
        __syncthreads();   // every wave waited on its own async loads -> buffer ready

        const float (*bqI)[LDS_STRIDE] = ldsbuf[cur][0];
        const float (*bqJ)[LDS_STRIDE] = ldsbuf[cur][1];
        const float (*bkI)[LDS_STRIDE] = ldsbuf[cur][2];
        const float (*bkJ)[LDS_STRIDE] = ldsbuf[cur][3];

#pragma unroll
        for (int kk = 0; kk < KB; kk += 4) {
            // A fragment (16x4 f32): lane -> (M = ln, K = kk + 2*hf + {0,1})
            float2 fa_q = *(const float2*)&bqI[wave * 16 + ln][kk + 2 * hf];
            float2 fa_k = *(const float2*)&bkI[wave * 16 + ln][kk + 2 * hf];
            v2f aQ = {fa_q.x, fa_q.y};
            v2f aK = {fa_k.x, fa_k.y};
#pragma unroll
            for (int tt = 0; tt < 8; ++tt) {
                // B fragment (4x16 f32): lane -> (K = kk + hf + 2*v, N = ln); B[k][n] = Xj[n][k]
                int n = tt * 16 + ln;
                v2f bQ = {bqJ[n][kk + hf], bqJ[n][kk + hf + 2]};
                v2f bK = {bkJ[n][kk + hf], bkJ[n][kk + hf + 2]};
                accq[tt] = __builtin_amdgcn_wmma_f32_16x16x4_f32(
                    false, aQ, false, bQ, (short)0, accq[tt], false, false);
                acck[tt] = __builtin_amdgcn_wmma_f32_16x16x4_f32(
                    false, aK, false, bK, (short)0, acck[tt], false, false);
            }
        }
        __syncthreads();   // all waves done reading before this buffer is overwritten
    }

    // Epilogue: cosine scaling, diagonal mask, squared difference.
    // C layout: VGPR r, lanes 0-15 -> M=r, lanes 16-31 -> M=r+8; N = ln.
    const int rowBase = Ibase + wave * 16 + hf * 8;
    float arow_q[8], srow_q[8], arow_k[8], srow_k[8];
#pragma unroll
    for (int rr = 0; rr < 8; ++rr) {
        int row = rowBase + rr;
        arow_q[rr] = aq[row]; srow_q[rr] = sq[row];
        arow_k[rr] = ak[row]; srow_k[rr] = sk[row];
    }
    const float offw = (bi == bj) ? 1.0f : 2.0f;   // off-diagonal blocks count twice
    float wsum = 0.0f;
#pragma unroll
    for (int tt = 0; tt < 8; ++tt) {
        int col = Jbase + tt * 16 + ln;
        float acol_q = aq[col], scol_q = sq[col];
        float acol_k = ak[col], scol_k = sk[col];
#pragma unroll
        for (int rr = 0; rr < 8; ++rr) {
            int row = rowBase + rr;
            float Sq = (accq[tt][rr] * arow_q[rr] * acol_q) / fmaxf(srow_q[rr] * scol_q, EPS_COS);
            float Sk = (acck[tt][rr] * arow_k[rr] * acol_k) / fmaxf(srow_k[rr] * scol_k, EPS_COS);
            float d  = Sq - Sk;
            float w  = (row == col) ? 0.0f : offw;
            wsum += w * d * d;
        }
    }

    // Deterministic block reduction: shuffle within wave, fixed-order across waves.
#pragma unroll
    for (int off = 16; off > 0; off >>= 1) wsum += __shfl_xor(wsum, off, 32);
    if (lane == 0) wave_sums[wave] = wsum;
    __syncthreads();
    if (tid == 0) {
        float bsum = 0.0f;
#pragma unroll
        for (int w = 0; w < 8; ++w) bsum += wave_sums[w];
        partial[blockIdx.x] = bsum;
    }
}

// ---------------------------------------------------------------------------
// Kernel 3: fixed-order final reduction + scale. Single block.
// ---------------------------------------------------------------------------
__global__ __launch_bounds__(256) void final_reduce_kernel(const float* __restrict__ partial,
                                                           int nparts,
                                                           float* __restrict__ out,
                                                           float scale) {
    __shared__ float sm[256];
    float s = 0.0f;
    for (int i = threadIdx.x; i < nparts; i += 256) s += partial[i];
    sm[threadIdx.x] = s;
    __syncthreads();
    for (int off = 128; off > 0; off >>= 1) {
        if (threadIdx.x < off) sm[threadIdx.x] += sm[threadIdx.x + off];
        __syncthreads();
    }
    if (threadIdx.x == 0) out[0] = sm[0] * scale;
}

extern "C" void kernel_launch(void* const* d_in, const int* in_sizes, int n_in,
                              void* d_out, int out_size, void* d_ws, size_t ws_size,
                              hipStream_t stream) {
    const float* fq = (const float*)d_in[0];
    const float* fk = (const float*)d_in[1];
    const int D = 1024;
    const int N = in_sizes[0] / D;          // 4096

    float* ws = (float*)d_ws;
    float* aq = ws;                         // [N]
    float* sq = ws + N;                     // [N]
    float* ak = ws + 2 * N;                 // [N]
    float* sk = ws + 3 * N;                 // [N]
    float* partial = ws + 4 * N;            // [nblk*(nblk+1)/2]

    const int nblk = N / BLK;               // 32
    const int ntri = nblk * (nblk + 1) / 2; // 528

    row_norms_kernel<<<N / 8, 256, 0, stream>>>(fq, aq, sq, D);
    row_norms_kernel<<<N / 8, 256, 0, stream>>>(fk, ak, sk, D);
    gram_diff_kernel<<<ntri, 256, 0, stream>>>(fq, fk, aq, sq, ak, sk, partial, N, D, nblk);
    final_reduce_kernel<<<1, 256, 0, stream>>>(partial, ntri, (float*)d_out,
                                               1.0f / ((float)N * (float)(N - 1)));
}